// Column_82136954569126
// MI455X (gfx1250) — compile-verified
//
#include <hip/hip_runtime.h>

// ---------------------------------------------------------------------------
// Problem constants (from the reference)
// ---------------------------------------------------------------------------
#define TSTEPS 64        // timesteps (GEMM M... used as WMMA N here)
#define NFEAT  2048      // features  (WMMA M)
#define KDIM   65536     // reduction (1*256*256)
#define SPLITS 16        // split-K factor
#define KSPLIT (KDIM / SPLITS)   // 4096 K per split
#define FTHRESH 16384.0f
#define KWTA   16

typedef float v2f __attribute__((ext_vector_type(2)));
typedef float v8f __attribute__((ext_vector_type(8)));

// ---------------------------------------------------------------------------
// Workspace layout (float slots)
// ---------------------------------------------------------------------------
#define WS_PARTIAL 0                                         // SPLITS*NFEAT*TSTEPS
#define WS_POT     (SPLITS * NFEAT * TSTEPS)                 // NFEAT*TSTEPS, layout [feat][t]
#define WS_NSPK    (WS_POT + NFEAT * TSTEPS)                 // NFEAT
#define WS_FPOT    (WS_NSPK + NFEAT)                         // NFEAT
#define WS_COEF    (WS_FPOT + NFEAT)                         // NFEAT
#define WS_MAXV    (WS_COEF + NFEAT)                         // 1 (uint bits of max trunc pot)

// ---------------------------------------------------------------------------
// Kernel 1: split-K GEMM with V_WMMA_F32_16X16X4_F32.
//   WMMA A = weight tile (16 feat x 4 k), WMMA B = rec_field tile (4 k x 16 t),
//   D/C    = 16 feat x 16 t fp32 accumulator (8 VGPRs).
// ISA fragment layouts (CDNA5 ISA, 32-bit A 16x4 / C 16x16):
//   A: lane(0-15) holds [row=lane, K=k0+0..1], lane(16-31) holds [K=k0+2..3]
//   B: mirrored K striping, N=lane%16 -> one contiguous float2 per lane each.
// Both operands therefore load as single global_load_b64 of contiguous K.
// ---------------------------------------------------------------------------
__global__ __launch_bounds__(128)
void Column_gemm_wmma(const float* __restrict__ rec,   // (64, 65536) K-contiguous
                      const float* __restrict__ wgt,   // (2048, 65536) K-contiguous
                      float* __restrict__ partial)     // (SPLITS, NFEAT, TSTEPS)
{
    const int lane  = threadIdx.x & 31;
    const int wave  = threadIdx.x >> 5;     // 4 waves: one 16-timestep tile each
    const int r     = lane & 15;            // row within 16-tile
    const int h     = lane >> 4;            // K half-select (0 -> k0+0..1, 1 -> k0+2..3)
    const int featBase = blockIdx.x * 16;   // 128 feature tiles
    const int split    = blockIdx.y;        // 16 K-splits
    const long kBegin  = (long)split * KSPLIT;
    const int  tBase   = wave * 16;

    const float* aptr = wgt + (size_t)(featBase + r) * KDIM + kBegin + h * 2;
    const float* bptr = rec + (size_t)(tBase    + r) * KDIM + kBegin + h * 2;

    v8f acc = {0.f, 0.f, 0.f, 0.f, 0.f, 0.f, 0.f, 0.f};

    // 64 chunks x 16 WMMAs; prefetch the streamed weight 1KB ahead per lane.
    for (int kc = 0; kc < KSPLIT; kc += 64) {
        __builtin_prefetch(aptr + kc + 256, 0, 1);
#pragma unroll
        for (int k = kc; k < kc + 64; k += 4) {
            v2f a = *(const v2f*)(aptr + k);
            v2f b = *(const v2f*)(bptr + k);
            acc = __builtin_amdgcn_wmma_f32_16x16x4_f32(
                /*neg_a=*/false, a, /*neg_b=*/false, b,
                /*c_mod=*/(short)0, acc, /*reuse_a=*/false, /*reuse_b=*/false);
        }
    }

    // D layout: VGPR i -> feat = featBase + i + h*8, t = tBase + (lane%16)
    size_t outBase = ((size_t)split * NFEAT + featBase + h * 8) * TSTEPS + tBase + r;
#pragma unroll
    for (int i = 0; i < 8; ++i)
        partial[outBase + (size_t)i * TSTEPS] = acc[i];
}

// ---------------------------------------------------------------------------
// Kernel 2: sum split-K partials, apply firing threshold -> pot[feat][t]
// ---------------------------------------------------------------------------
__global__ __launch_bounds__(256)
void Column_reduce_thresh(const float* __restrict__ partial,
                          float* __restrict__ pot,
                          unsigned* __restrict__ maxv)
{
    int p = blockIdx.x * blockDim.x + threadIdx.x;    // 0 .. NFEAT*TSTEPS-1
    float s = 0.f;
#pragma unroll
    for (int sp = 0; sp < SPLITS; ++sp)
        s += partial[(size_t)sp * (NFEAT * TSTEPS) + p];
    pot[p] = (s > FTHRESH) ? s : 0.0f;
    if (p == 0) maxv[0] = 0u;                         // init for kernel 3's atomicMax
}

// ---------------------------------------------------------------------------
// Kernel 3: per-feature spike count + first-spike potential + global trunc max
// first_idx = clamp(T - n_spk, 0, T-1); contrib to global max only if n_spk>0.
// Non-negative floats compare identically as unsigned bit patterns -> atomicMax
// is order-independent (deterministic).
// ---------------------------------------------------------------------------
__global__ __launch_bounds__(256)
void Column_feature_stats(const float* __restrict__ pot,
                          float* __restrict__ nspk,
                          float* __restrict__ fpot,
                          unsigned* __restrict__ maxv)
{
    int k = blockIdx.x * blockDim.x + threadIdx.x;    // 0 .. NFEAT-1
    const float* p = pot + (size_t)k * TSTEPS;
    int n = 0;
#pragma unroll 8
    for (int t = 0; t < TSTEPS; ++t)
        n += (p[t] > 0.0f) ? 1 : 0;
    int fi = TSTEPS - n;
    if (fi > TSTEPS - 1) fi = TSTEPS - 1;
    float fp = p[fi];
    nspk[k] = (float)n;
    fpot[k] = fp;
    float contrib = (n > 0) ? fp : 0.0f;              // >= 0 always
    atomicMax(maxv, (unsigned)__float_as_int(contrib));
}

// ---------------------------------------------------------------------------
// Kernel 4 (single WG): total[k] = n_spk * (first_pot + maxtrunc*T), then 16
// rounds of argmax (ties -> lowest index, matching lax.top_k) with zeroing.
// ---------------------------------------------------------------------------
__global__ __launch_bounds__(256)
void Column_select_winners(const float* __restrict__ nspk,
                           const float* __restrict__ fpot,
                           const unsigned* __restrict__ maxv,
                           float* __restrict__ coef)
{
    __shared__ float tot[NFEAT];
    __shared__ float rv[256];
    __shared__ int   ri[256];
    const int tid = threadIdx.x;

    const float v = __int_as_float((int)maxv[0]) * (float)TSTEPS;
    for (int k = tid; k < NFEAT; k += 256) {
        tot[k]  = nspk[k] * (fpot[k] + v);
        coef[k] = 0.0f;
    }
    __syncthreads();

    for (int iter = 0; iter < KWTA; ++iter) {
        float bv = -1.0f; int bi = 0;
        for (int k = tid; k < NFEAT; k += 256) {      // ascending k, strict > keeps lowest idx
            float x = tot[k];
            if (x > bv) { bv = x; bi = k; }
        }
        rv[tid] = bv; ri[tid] = bi;
        __syncthreads();
        for (int off = 128; off > 0; off >>= 1) {
            if (tid < off) {
                float ov = rv[tid + off]; int oi = ri[tid + off];
                if (ov > rv[tid] || (ov == rv[tid] && oi < ri[tid])) {
                    rv[tid] = ov; ri[tid] = oi;
                }
            }
            __syncthreads();
        }
        if (tid == 0) {
            int w = ri[0];
            if (rv[0] > 0.0f) coef[w] = 1.0f;         // valid = top_val > 0
            tot[w] = -1.0f;                           // inhibit winner (radius 0)
        }
        __syncthreads();
    }
}

// ---------------------------------------------------------------------------
// Kernel 5: out[t,k,0,0] = sign(pot[k][t] * coef[k]) in (T, K) order
// ---------------------------------------------------------------------------
__global__ __launch_bounds__(256)
void Column_write_out(const float* __restrict__ pot,
                      const float* __restrict__ coef,
                      float* __restrict__ out)
{
    int idx = blockIdx.x * blockDim.x + threadIdx.x;  // 0 .. TSTEPS*NFEAT-1
    int t = idx / NFEAT;
    int k = idx - t * NFEAT;
    out[idx] = (coef[k] > 0.0f && pot[(size_t)k * TSTEPS + t] > 0.0f) ? 1.0f : 0.0f;
}

// ---------------------------------------------------------------------------
extern "C" void kernel_launch(void* const* d_in, const int* in_sizes, int n_in,
                              void* d_out, int out_size, void* d_ws, size_t ws_size,
                              hipStream_t stream)
{
    const float* rec = (const float*)d_in[0];   // rec_field (64,1,256,256)
    const float* wgt = (const float*)d_in[1];   // weight    (2048,1,256,256)
    float* ws = (float*)d_ws;

    float*    partial = ws + WS_PARTIAL;
    float*    pot     = ws + WS_POT;
    float*    nspk    = ws + WS_NSPK;
    float*    fpot    = ws + WS_FPOT;
    float*    coef    = ws + WS_COEF;
    unsigned* maxv    = (unsigned*)(ws + WS_MAXV);

    dim3 g1(NFEAT / 16, SPLITS);                // 128 x 16 = 2048 WGs, 4 waves each
    Column_gemm_wmma<<<g1, 128, 0, stream>>>(rec, wgt, partial);

    Column_reduce_thresh<<<(NFEAT * TSTEPS) / 256, 256, 0, stream>>>(partial, pot, maxv);
    Column_feature_stats<<<NFEAT / 256, 256, 0, stream>>>(pot, nspk, fpot, maxv);
    Column_select_winners<<<1, 256, 0, stream>>>(nspk, fpot, maxv, coef);
    Column_write_out<<<(NFEAT * TSTEPS) / 256, 256, 0, stream>>>(pot, coef, (float*)d_out);
}